// HealpixEncoder_32693291057254
// MI455X (gfx1250) — compile-verified
//
#include <hip/hip_runtime.h>

typedef __attribute__((ext_vector_type(2))) float v2f;
typedef __attribute__((ext_vector_type(8))) float v8f;

#define LN_EPS 1e-5f

static inline long cdiv(long a, long b) { return (a + b - 1) / b; }

// ---------------------------------------------------------------------------
// Copy stage input (B,P,C) into slice 0 of xk (B,P,4C)
// ---------------------------------------------------------------------------
__global__ void cheb_copy_t0(const float* __restrict__ src, float* __restrict__ xk,
                             int C, long total) {
  long idx = (long)blockIdx.x * blockDim.x + threadIdx.x;
  if (idx >= total) return;
  int c = (int)(idx % C);
  long bp = idx / C;
  xk[bp * (long)(4 * C) + c] = src[idx];
}

// ---------------------------------------------------------------------------
// One Chebyshev sparse pass:
//   dst[b,p,c] = scale * sum_d vals[p,d] * src[b, cols[p,d], c]  (- sub[b,p,c])
// src/sub/dst addressed with (batchStride, rowStride, chanOffset) so xk slices
// (row stride 4C) and plain (B,P,C) tensors both work.
// ---------------------------------------------------------------------------
__global__ void cheb_lmul(const int* __restrict__ cols, const float* __restrict__ vals,
                          const float* __restrict__ src, long sB, int sR, int sO,
                          const float* __restrict__ sub, long uB, int uR, int uO,
                          float* __restrict__ dst, long dB, int dR, int dO,
                          int P, int C, float scale, int hasSub, long total) {
  long idx = (long)blockIdx.x * blockDim.x + threadIdx.x;
  if (idx >= total) return;
  int c = (int)(idx % C);
  long t = idx / C;
  int p = (int)(t % P);
  int b = (int)(t / P);
  const int*   cp = cols + (long)p * 8;
  const float* vp = vals + (long)p * 8;
  float acc = 0.f;
#pragma unroll
  for (int d = 0; d < 8; ++d) {
    int q = cp[d];
    acc += vp[d] * src[(long)b * sB + (long)q * sR + sO + c];
  }
  acc *= scale;
  if (hasSub) acc -= sub[(long)b * uB + (long)p * uR + uO + c];
  dst[(long)b * dB + (long)p * dR + dO + c] = acc;
}

// ---------------------------------------------------------------------------
// fp32 GEMM via V_WMMA_F32_16X16X4_F32 with N-register-blocking.
// One wave32 computes a full 16-row x (NT*16)-col stripe: the A fragment is
// loaded once per K-step and reused for NT WMMAs (A is the large operand —
// up to 201MB — while W is KB-sized and cache-hot).
//   A: (M, Kd) row-major (xk),  W: (Kd, N=NT*16) row-major,  out = A@W + bias
// A-frag layout (ISA 7.12.2, 32-bit A 16x4): lanes 0-15 / 16-31 both hold
// M=0..15; VGPR0 = K0|K2, VGPR1 = K1|K3 across lane halves. B mirrors with
// N striped across lanes. C/D: VGPR r -> M=r (lanes 0-15), M=r+8 (16-31).
// ---------------------------------------------------------------------------
template <int NT>
__global__ __launch_bounds__(32)
void gemm_bias_wmma(const float* __restrict__ A, const float* __restrict__ W,
                    const float* __restrict__ bias, float* __restrict__ out,
                    int Kd) {
  const int N = NT * 16;
  int lane = threadIdx.x;
  int lm   = lane & 15;
  int half = lane >> 4;
  long m0 = (long)blockIdx.x * 16;

  v8f acc[NT];
#pragma unroll
  for (int t = 0; t < NT; ++t) {
    float bv = bias[t * 16 + lm];
#pragma unroll
    for (int r = 0; r < 8; ++r) acc[t][r] = bv;
  }

  const float* arow = A + (m0 + lm) * (long)Kd;
  for (int k = 0; k < Kd; k += 4) {
    int ka = k + 2 * half;                       // K0/K2 in v0, K1/K3 in v1
    v2f a = *(const v2f*)(arow + ka);            // one b64 load, reused NT times
    const float* w0 = W + (long)ka * N + lm;
    const float* w1 = w0 + N;
#pragma unroll
    for (int t = 0; t < NT; ++t) {
      v2f b; b.x = w0[t * 16]; b.y = w1[t * 16];
      acc[t] = __builtin_amdgcn_wmma_f32_16x16x4_f32(false, a, false, b,
                                                     (short)0, acc[t], false, false);
    }
  }

#pragma unroll
  for (int t = 0; t < NT; ++t) {
#pragma unroll
    for (int r = 0; r < 8; ++r) {
      long row = m0 + r + 8 * half;
      out[row * (long)N + t * 16 + lm] = acc[t][r];
    }
  }
}

// ---------------------------------------------------------------------------
// LayerNorm (channel dim) + ReLU + average-pool over `pool` consecutive rows.
// One block (C threads) per output row.  pool=1 => no pooling.
// ---------------------------------------------------------------------------
__global__ void ln_relu_pool(const float* __restrict__ h, const float* __restrict__ g,
                             const float* __restrict__ bt, float* __restrict__ out,
                             int C, int pool) {
  extern __shared__ float sm[];
  int c = threadIdx.x;
  long orow = blockIdx.x;
  float invC = 1.f / (float)C;
  float acc = 0.f;
  for (int j = 0; j < pool; ++j) {
    long row = orow * pool + j;
    float v = h[row * (long)C + c];
    sm[c] = v; __syncthreads();
    for (int s = C >> 1; s > 0; s >>= 1) { if (c < s) sm[c] += sm[c + s]; __syncthreads(); }
    float mu = sm[0] * invC; __syncthreads();
    float d = v - mu;
    sm[c] = d * d; __syncthreads();
    for (int s = C >> 1; s > 0; s >>= 1) { if (c < s) sm[c] += sm[c + s]; __syncthreads(); }
    float var = sm[0] * invC; __syncthreads();
    float y = g[c] * d / sqrtf(var + LN_EPS) + bt[c];
    acc += fmaxf(y, 0.f);
  }
  out[orow * (long)C + c] = acc / (float)pool;
}

// ---------------------------------------------------------------------------
// Deterministic global mean pool over pixels (two-stage, no fp atomics)
// ---------------------------------------------------------------------------
__global__ void colsum_partial(const float* __restrict__ h, float* __restrict__ part,
                               int P, int C, int chunk) {
  int b = blockIdx.x, k = blockIdx.y, c = threadIdx.x;
  const float* base = h + ((long)b * P + (long)k * chunk) * C + c;
  float s = 0.f;
  for (int i = 0; i < chunk; ++i) s += base[(long)i * C];
  part[((long)b * gridDim.y + k) * C + c] = s;
}

__global__ void colsum_final(const float* __restrict__ part, float* __restrict__ z,
                             int nchunk, int C, float inv) {
  int b = blockIdx.x, c = threadIdx.x;
  float s = 0.f;
  for (int k = 0; k < nchunk; ++k) s += part[((long)b * nchunk + k) * C + c];
  z[(long)b * C + c] = s * inv;
}

// MLP layer: o[b,j] = (relu?)(sum_c z[b,c]*W[c,j] + bias[j]);  grid=B, block=Cout
__global__ void mlp_layer(const float* __restrict__ z, const float* __restrict__ W,
                          const float* __restrict__ bias, float* __restrict__ o,
                          int Cin, int Cout, int relu) {
  int b = blockIdx.x, j = threadIdx.x;
  float s = bias[j];
  for (int c = 0; c < Cin; ++c) s += z[(long)b * Cin + c] * W[(long)c * Cout + j];
  if (relu) s = fmaxf(s, 0.f);
  o[(long)b * Cout + j] = s;
}

// ---------------------------------------------------------------------------
// Host-side stage driver
// ---------------------------------------------------------------------------
static void run_stage(const float* xin, const int* cols, const float* vals,
                      const float* W, const float* bias, const float* g, const float* bt,
                      float* xk, float* h, float* outp,
                      int P, int Cin, int Cout, int pool, hipStream_t stream) {
  const int Bsz = 4;
  long M = (long)Bsz * P;
  int Kd = 4 * Cin;
  long totalC = M * Cin;
  int TB = 256;
  long nb = cdiv(totalC, TB);
  long xkB = (long)P * Kd;   // xk batch stride
  long inB = (long)P * Cin;  // plain (B,P,C) batch stride

  cheb_copy_t0<<<nb, TB, 0, stream>>>(xin, xk, Cin, totalC);
  // t1 = L t0
  cheb_lmul<<<nb, TB, 0, stream>>>(cols, vals, xin, inB, Cin, 0,
                                   nullptr, 0, 0, 0,
                                   xk, xkB, Kd, Cin, P, Cin, 1.f, 0, totalC);
  // t2 = 2 L t1 - t0
  cheb_lmul<<<nb, TB, 0, stream>>>(cols, vals, xk, xkB, Kd, Cin,
                                   xk, xkB, Kd, 0,
                                   xk, xkB, Kd, 2 * Cin, P, Cin, 2.f, 1, totalC);
  // t3 = 2 L t2 - t1
  cheb_lmul<<<nb, TB, 0, stream>>>(cols, vals, xk, xkB, Kd, 2 * Cin,
                                   xk, xkB, Kd, Cin,
                                   xk, xkB, Kd, 3 * Cin, P, Cin, 2.f, 1, totalC);
  // GEMM + bias (WMMA fp32, N-register-blocked: one wave per 16-row stripe)
  unsigned mtiles = (unsigned)(M / 16);
  if (Cout == 64) {
    gemm_bias_wmma<4><<<mtiles, 32, 0, stream>>>(xk, W, bias, h, Kd);
  } else if (Cout == 128) {
    gemm_bias_wmma<8><<<mtiles, 32, 0, stream>>>(xk, W, bias, h, Kd);
  } else {
    gemm_bias_wmma<16><<<mtiles, 32, 0, stream>>>(xk, W, bias, h, Kd);
  }
  // LN + ReLU + pool
  long orows = M / pool;
  ln_relu_pool<<<(unsigned)orows, Cout, Cout * sizeof(float), stream>>>(h, g, bt, outp, Cout, pool);
}

extern "C" void kernel_launch(void* const* d_in, const int* in_sizes, int n_in,
                              void* d_out, int out_size, void* d_ws, size_t ws_size,
                              hipStream_t stream) {
  (void)in_sizes; (void)n_in; (void)out_size; (void)ws_size;
  const float* x     = (const float*)d_in[0];
  const int*   cols1 = (const int*)  d_in[1];
  const float* vals1 = (const float*)d_in[2];
  const int*   cols2 = (const int*)  d_in[3];
  const float* vals2 = (const float*)d_in[4];
  const int*   cols3 = (const int*)  d_in[5];
  const float* vals3 = (const float*)d_in[6];
  const float* W1 = (const float*)d_in[7],  *b1 = (const float*)d_in[8];
  const float* g1 = (const float*)d_in[9],  *bt1 = (const float*)d_in[10];
  const float* W2 = (const float*)d_in[11], *b2 = (const float*)d_in[12];
  const float* g2 = (const float*)d_in[13], *bt2 = (const float*)d_in[14];
  const float* W3 = (const float*)d_in[15], *b3 = (const float*)d_in[16];
  const float* g3 = (const float*)d_in[17], *bt3 = (const float*)d_in[18];
  const float* mW1 = (const float*)d_in[19], *mb1 = (const float*)d_in[20];
  const float* mW2 = (const float*)d_in[21], *mb2 = (const float*)d_in[22];

  const int P0 = 196608, P1 = 49152, P2 = 12288;

  // Workspace regions (floats). Peak live: A(201MB) + B(201MB) + C(50MB) + tail.
  float* ws = (float*)d_ws;
  float* rA = ws;                      // xk region   (max 50,331,648 floats)
  float* rB = ws + 50331648L;          // h region    (max 50,331,648 floats)
  float* rC = ws + 100663296L;         // pooled/ln3  (max 12,582,912 floats)
  float* rPart = ws + 113246208L;      // 4*48*256 partial sums
  float* rZ    = rPart + 49152;        // 4*256
  float* rHid  = rZ + 1024;            // 4*512

  // Stage 1: (4,196608,16) -> pooled (4,49152,64)
  run_stage(x,  cols1, vals1, W1, b1, g1, bt1, rA, rB, rC, P0, 16,  64, 4, stream);
  // Stage 2: (4,49152,64) -> pooled (4,12288,128)
  run_stage(rC, cols2, vals2, W2, b2, g2, bt2, rA, rB, rC, P1, 64, 128, 4, stream);
  // Stage 3: (4,12288,128) -> (4,12288,256), no pooling (written back over rC)
  run_stage(rC, cols3, vals3, W3, b3, g3, bt3, rA, rB, rC, P2, 128, 256, 1, stream);

  // Global mean pool over pixels (deterministic two-stage reduction)
  const int NCH = 48, CHUNK = P2 / NCH;  // 48 chunks of 256 pixels
  dim3 gp(4, NCH);
  colsum_partial<<<gp, 256, 0, stream>>>(rC, rPart, P2, 256, CHUNK);
  colsum_final<<<4, 256, 0, stream>>>(rPart, rZ, NCH, 256, 1.f / (float)P2);

  // MLP head: 256 -> 512 (ReLU) -> 256
  mlp_layer<<<4, 512, 0, stream>>>(rZ, mW1, mb1, rHid, 256, 512, 1);
  mlp_layer<<<4, 256, 0, stream>>>(rHid, mW2, mb2, (float*)d_out, 512, 256, 0);
}